// ATRM_73581379715509
// MI455X (gfx1250) — compile-verified
//
#include <hip/hip_runtime.h>
#include <math.h>

#define DIMX     512
#define NTOK     256
#define HIDDEN   102
#define KEEPED   51
#define NKEEP    128
#define OUTROWS  52

#define SELP 68    // 64-wide sel chunk, pitch 68 (stride%64==4 -> conflict-free row gather)
#define W1P  113   // 112-wide W1 chunk, pitch 113
#define H1P  106   // 104-wide h1 (K padded), pitch 106 (stride%64==42, 16 rows distinct banks)
#define WTP  130   // 128-wide wT,            pitch 130 (stride%64==2)

typedef __attribute__((ext_vector_type(2))) float v2f;
typedef __attribute__((ext_vector_type(8))) float v8f;

__device__ __forceinline__ v8f wmma4(v2f a, v2f b, v8f c) {
  // D = A(16x4) * B(4x16) + C(16x16), fp32 matrix pipe on gfx1250
  return __builtin_amdgcn_wmma_f32_16x16x4_f32(false, a, false, b, (short)0, c,
                                               false, false);
}

__device__ __forceinline__ float wredSum(float v) {
#pragma unroll
  for (int o = 16; o > 0; o >>= 1) v += __shfl_xor(v, o, 32);
  return v;
}
__device__ __forceinline__ float wredMax(float v) {
#pragma unroll
  for (int o = 16; o > 0; o >>= 1) v = fmaxf(v, __shfl_xor(v, o, 32));
  return v;
}

__global__ __launch_bounds__(256) void atrm_fused(
    const float* __restrict__ x, const float* __restrict__ ca,
    const float* __restrict__ ln_g, const float* __restrict__ ln_b,
    const float* __restrict__ W1, const float* __restrict__ b1,
    const float* __restrict__ W2, const float* __restrict__ b2,
    const float* __restrict__ scale, float* __restrict__ out) {
  __shared__ float s_glo[DIMX];
  __shared__ float s_g[DIMX];
  __shared__ float s_bb[DIMX];
  __shared__ float s_score[NTOK];
  __shared__ int   s_idx[NTOK];
  __shared__ float s_nonw[NKEEP];
  __shared__ float s_mu[NKEEP];
  __shared__ float s_rstd[NKEEP];
  __shared__ float s_red[64];
  __shared__ float s_selt[NKEEP * SELP];   // 64-col chunk of gathered sel rows
  __shared__ float s_w1t[64 * W1P];        // 64-row chunk of W1 (N padded to 112)
  __shared__ float s_h1[NKEEP * H1P];      // gelu(LN(sel)@W1+b1), K padded to 104
  __shared__ float s_wT[64 * WTP];         // transposed slot weights (64x128)

  const int tid  = threadIdx.x;
  const int lane = tid & 31;
  const int wid  = tid >> 5;
  const int half = lane >> 4;
  const int l16  = lane & 15;
  const int b    = blockIdx.x;
  const float* xb = x + (size_t)b * NTOK * DIMX;

  // ---- stage LN params ----
  s_g[tid] = ln_g[tid];       s_g[tid + 256] = ln_g[tid + 256];
  s_bb[tid] = ln_b[tid];      s_bb[tid + 256] = ln_b[tid + 256];

  // ---- phase 1a: mean over tokens, then glo = l2norm(mean) ----
  float m0 = 0.f, m1 = 0.f;
  for (int t = 0; t < NTOK; ++t) {
    m0 += xb[t * DIMX + tid];
    m1 += xb[t * DIMX + tid + 256];
  }
  m0 *= (1.f / NTOK);
  m1 *= (1.f / NTOK);
  s_glo[tid] = m0;
  s_glo[tid + 256] = m1;
  float ss = wredSum(m0 * m0 + m1 * m1);
  if (lane == 0) s_red[wid] = ss;
  __syncthreads();
  if (tid == 0) {
    float tot = 0.f;
#pragma unroll
    for (int i = 0; i < 8; ++i) tot += s_red[i];
    s_red[32] = 1.f / fmaxf(sqrtf(tot), 1e-12f);
  }
  __syncthreads();
  {
    float inv = s_red[32];
    s_glo[tid] *= inv;
    s_glo[tid + 256] *= inv;
  }
  __syncthreads();

  // ---- phase 1b: scores = dot(glo, x_t)/||x_t|| + cross_att ----
  for (int t = wid; t < NTOK; t += 8) {
    float dt = 0.f, sq = 0.f;
#pragma unroll
    for (int j = 0; j < 16; ++j) {
      int c = j * 32 + lane;
      float v = xb[t * DIMX + c];
      dt += s_glo[c] * v;
      sq += v * v;
    }
    dt = wredSum(dt);
    sq = wredSum(sq);
    if (lane == 0) {
      s_score[t] = dt / fmaxf(sqrtf(sq), 1e-12f) + ca[b * NTOK + t];
      s_idx[t] = t;
    }
  }
  __syncthreads();

  // ---- phase 2: bitonic sort (descending) of 256 (score, idx) ----
  for (int k = 2; k <= NTOK; k <<= 1) {
    for (int j = k >> 1; j > 0; j >>= 1) {
      int ixj = tid ^ j;
      if (ixj > tid) {
        bool dir = ((tid & k) == 0);  // descending segments
        float a = s_score[tid], c = s_score[ixj];
        bool sw = dir ? (a < c) : (a > c);
        if (sw) {
          s_score[tid] = c; s_score[ixj] = a;
          int t1 = s_idx[tid]; s_idx[tid] = s_idx[ixj]; s_idx[ixj] = t1;
        }
      }
      __syncthreads();
    }
  }

  // ---- phase 3: softmax over dropped-token scores (wave 0) ----
  if (wid == 0) {
    float v0 = s_score[NKEEP + lane];
    float v1 = s_score[NKEEP + lane + 32];
    float v2 = s_score[NKEEP + lane + 64];
    float v3 = s_score[NKEEP + lane + 96];
    float mx = wredMax(fmaxf(fmaxf(v0, v1), fmaxf(v2, v3)));
    float e0 = expf(v0 - mx), e1 = expf(v1 - mx);
    float e2 = expf(v2 - mx), e3 = expf(v3 - mx);
    float sm = wredSum(e0 + e1 + e2 + e3);
    float inv = 1.f / sm;
    s_nonw[lane] = e0 * inv;
    s_nonw[lane + 32] = e1 * inv;
    s_nonw[lane + 64] = e2 * inv;
    s_nonw[lane + 96] = e3 * inv;
  }
  __syncthreads();

  // ---- phase 4: extra = sum_j nonw[j] * x[non_idx[j]] ----
  {
    float e0 = 0.f, e1 = 0.f;
    for (int j = 0; j < NKEEP; ++j) {
      int t = s_idx[NKEEP + j];
      float w = s_nonw[j];
      e0 += w * xb[t * DIMX + tid];
      e1 += w * xb[t * DIMX + tid + 256];
    }
    float* orow = out + ((size_t)b * OUTROWS + 51) * DIMX;
    orow[tid] = e0;
    orow[tid + 256] = e1;
  }

  // ---- phase 5: LN stats per kept row ----
  for (int rr = wid; rr < NKEEP; rr += 8) {
    int t = s_idx[rr];
    float sm = 0.f, sq = 0.f;
#pragma unroll
    for (int j = 0; j < 16; ++j) {
      float v = xb[t * DIMX + j * 32 + lane];
      sm += v; sq += v * v;
    }
    sm = wredSum(sm);
    sq = wredSum(sq);
    if (lane == 0) {
      float mu = sm * (1.f / DIMX);
      float var = sq * (1.f / DIMX) - mu * mu;
      s_mu[rr] = mu;
      s_rstd[rr] = rsqrtf(fmaxf(var, 0.f) + 1e-5f);
    }
  }
  __syncthreads();

  // ---- GEMM1: h1 = gelu(LN(sel) @ W1 + b1)  (M=128, N=102->112, K=512) ----
  {
    const int m = wid;  // M-tile per wave
    v8f acc[7];
#pragma unroll
    for (int nt = 0; nt < 7; ++nt) {
      int col = nt * 16 + l16;
      float bias = (col < HIDDEN) ? b1[col] : 0.f;
#pragma unroll
      for (int r = 0; r < 8; ++r) acc[nt][r] = bias;
    }
    for (int ch = 0; ch < 8; ++ch) {
      __syncthreads();
      // stage 16 own sel rows (64 cols) : lane loads float2
#pragma unroll
      for (int rloc = 0; rloc < 16; ++rloc) {
        int row = m * 16 + rloc;
        int tok = s_idx[row];
        float2 v = *reinterpret_cast<const float2*>(
            xb + tok * DIMX + ch * 64 + 2 * lane);
        s_selt[row * SELP + 2 * lane] = v.x;
        s_selt[row * SELP + 2 * lane + 1] = v.y;
      }
      // cooperative stage of W1 chunk (64 x 112, zero-padded)
      for (int i = tid; i < 64 * 112; i += 256) {
        int r = i / 112, c = i - r * 112;
        s_w1t[r * W1P + c] =
            (c < HIDDEN) ? W1[(ch * 64 + r) * HIDDEN + c] : 0.f;
      }
      __syncthreads();
#pragma unroll
      for (int kk = 0; kk < 16; ++kk) {
        int kb = kk * 4 + 2 * half;
        int arow = m * 16 + l16;
        float va = s_selt[arow * SELP + kb];
        float vb = s_selt[arow * SELP + kb + 1];
        int d0 = ch * 64 + kb;
        float mu = s_mu[arow], rs = s_rstd[arow];
        v2f a;
        a.x = (va - mu) * rs * s_g[d0] + s_bb[d0];
        a.y = (vb - mu) * rs * s_g[d0 + 1] + s_bb[d0 + 1];
#pragma unroll
        for (int nt = 0; nt < 7; ++nt) {
          v2f bf;
          bf.x = s_w1t[kb * W1P + nt * 16 + l16];
          bf.y = s_w1t[(kb + 1) * W1P + nt * 16 + l16];
          acc[nt] = wmma4(a, bf, acc[nt]);
        }
      }
    }
    __syncthreads();
    // exact gelu + store to h1
#pragma unroll
    for (int nt = 0; nt < 7; ++nt) {
      int col = nt * 16 + l16;
      if (col < HIDDEN) {
#pragma unroll
        for (int r = 0; r < 8; ++r) {
          int row = m * 16 + r + 8 * half;
          float v = acc[nt][r];
          s_h1[row * H1P + col] = 0.5f * v * (1.f + erff(v * 0.70710678f));
        }
      }
    }
    if (tid < NKEEP) {  // zero K-padding cols 102..105
#pragma unroll
      for (int c = HIDDEN; c < H1P; ++c) s_h1[tid * H1P + c] = 0.f;
    }
  }
  __syncthreads();

  // ---- GEMM2: w = h1 @ W2 + b2  (M=128, N=51->64, K=102->104), store wT ----
  {
    const int m = wid;
    v8f acc2[4];
#pragma unroll
    for (int nt = 0; nt < 4; ++nt) {
      int col = nt * 16 + l16;
      float bias = (col < KEEPED) ? b2[col] : 0.f;
#pragma unroll
      for (int r = 0; r < 8; ++r) acc2[nt][r] = bias;
    }
#pragma unroll 2
    for (int kk = 0; kk < 26; ++kk) {
      int kb = kk * 4 + 2 * half;
      int arow = m * 16 + l16;
      v2f a;
      a.x = s_h1[arow * H1P + kb];
      a.y = s_h1[arow * H1P + kb + 1];
#pragma unroll
      for (int nt = 0; nt < 4; ++nt) {
        int col = nt * 16 + l16;
        v2f bf;
        bf.x = (kb < HIDDEN && col < KEEPED) ? W2[kb * KEEPED + col] : 0.f;
        bf.y = (kb + 1 < HIDDEN && col < KEEPED) ? W2[(kb + 1) * KEEPED + col] : 0.f;
        acc2[nt] = wmma4(a, bf, acc2[nt]);
      }
    }
#pragma unroll
    for (int nt = 0; nt < 4; ++nt) {
      int col = nt * 16 + l16;
#pragma unroll
      for (int r = 0; r < 8; ++r) {
        int row = m * 16 + r + 8 * half;
        s_wT[col * WTP + row] = acc2[nt][r];
      }
    }
  }
  __syncthreads();

  // ---- softmax over token dim for each slot row of wT (scale applied) ----
  {
    float scl = scale[0];
    for (int pi = 0; pi < 8; ++pi) {
      int p = wid * 8 + pi;
      float v0 = s_wT[p * WTP + lane] * scl;
      float v1 = s_wT[p * WTP + lane + 32] * scl;
      float v2 = s_wT[p * WTP + lane + 64] * scl;
      float v3 = s_wT[p * WTP + lane + 96] * scl;
      float mx = wredMax(fmaxf(fmaxf(v0, v1), fmaxf(v2, v3)));
      float e0 = expf(v0 - mx), e1 = expf(v1 - mx);
      float e2 = expf(v2 - mx), e3 = expf(v3 - mx);
      float sm = wredSum(e0 + e1 + e2 + e3);
      float inv = 1.f / sm;
      s_wT[p * WTP + lane] = e0 * inv;
      s_wT[p * WTP + lane + 32] = e1 * inv;
      s_wT[p * WTP + lane + 64] = e2 * inv;
      s_wT[p * WTP + lane + 96] = e3 * inv;
    }
  }
  __syncthreads();

  // ---- GEMM3: aggr = wT(51->64 x 128) @ sel(128 x 512) ----
  {
#pragma unroll
    for (int nt4 = 0; nt4 < 4; ++nt4) {
      int nt = wid * 4 + nt4;
      int ncol = nt * 16 + l16;
      v8f acc3[4];
#pragma unroll
      for (int mt = 0; mt < 4; ++mt)
#pragma unroll
        for (int r = 0; r < 8; ++r) acc3[mt][r] = 0.f;
#pragma unroll 4
      for (int kk = 0; kk < 32; ++kk) {
        int kb = kk * 4 + 2 * half;
        int tok0 = s_idx[kb];
        int tok1 = s_idx[kb + 1];
        v2f bf;
        bf.x = xb[tok0 * DIMX + ncol];
        bf.y = xb[tok1 * DIMX + ncol];
#pragma unroll
        for (int mt = 0; mt < 4; ++mt) {
          int arow = mt * 16 + l16;
          v2f a;
          a.x = s_wT[arow * WTP + kb];
          a.y = s_wT[arow * WTP + kb + 1];
          acc3[mt] = wmma4(a, bf, acc3[mt]);
        }
      }
#pragma unroll
      for (int mt = 0; mt < 4; ++mt) {
#pragma unroll
        for (int r = 0; r < 8; ++r) {
          int p = mt * 16 + r + 8 * half;
          if (p < KEEPED)
            out[((size_t)b * OUTROWS + p) * DIMX + ncol] = acc3[mt][r];
        }
      }
    }
  }
}

extern "C" void kernel_launch(void* const* d_in, const int* in_sizes, int n_in,
                              void* d_out, int out_size, void* d_ws,
                              size_t ws_size, hipStream_t stream) {
  const float* x    = (const float*)d_in[0];
  const float* ca   = (const float*)d_in[1];
  const float* ln_g = (const float*)d_in[2];
  const float* ln_b = (const float*)d_in[3];
  const float* W1   = (const float*)d_in[4];
  const float* b1   = (const float*)d_in[5];
  const float* W2   = (const float*)d_in[6];
  const float* b2   = (const float*)d_in[7];
  const float* sc   = (const float*)d_in[8];
  float* out = (float*)d_out;
  (void)in_sizes; (void)n_in; (void)out_size; (void)d_ws; (void)ws_size;
  atrm_fused<<<dim3(512), dim3(256), 0, stream>>>(x, ca, ln_g, ln_b, W1, b1,
                                                  W2, b2, sc, out);
}